// FastSearchModel_79946521247793
// MI455X (gfx1250) — compile-verified
//
#include <hip/hip_runtime.h>
#include <math.h>

typedef __attribute__((ext_vector_type(2))) float v2f;
typedef __attribute__((ext_vector_type(8))) float v8f;

#define EPSV 1e-6f
#define TM 128
#define TN 128
#define KC 32
#define LDT 36   // padded LDS row stride (floats); 36*4=144 bytes keeps float4 stores aligned

// ---------------------------------------------------------------- utilities
__global__ void zero_kernel(float* __restrict__ p, int n) {
  int i = blockIdx.x * blockDim.x + threadIdx.x;
  if (i < n) p[i] = 0.0f;
}

// out[row] = sum(x^2) + sgn*2*EPS*sum(x) + bias   (one wave32 per row)
__global__ void row_stats_kernel(const float* __restrict__ emb, float* __restrict__ out,
                                 int rows, int D, float sgn, float bias) {
  int row  = blockIdx.x * (blockDim.x >> 5) + (threadIdx.x >> 5);
  int lane = threadIdx.x & 31;
  if (row >= rows) return;
  const float* p = emb + (size_t)row * D;
  float s = 0.f, ss = 0.f;
  for (int k = lane; k < D; k += 32) { float x = p[k]; s += x; ss = fmaf(x, x, ss); }
#pragma unroll
  for (int o = 16; o > 0; o >>= 1) {
    s  += __shfl_down(s,  o, 32);
    ss += __shfl_down(ss, o, 32);
  }
  if (lane == 0) out[row] = ss + sgn * (2.0f * EPSV) * s + bias;
}

// --------------------------------------------- fused GEMM + sqrt + band sum
// C = src(8192x512) * subj^T(512x2048) via V_WMMA_F32_16X16X4_F32;
// epilogue: rn = sqrt(max(ca[r]+cb[j]-2*dot,0)), accumulated into dists[r-j].
// Double-buffered LDS with register staging: next chunk's global_load_b128s
// stay in flight across the current chunk's 64 WMMAs.
__global__ __launch_bounds__(256)
void dot_diag_kernel(const float* __restrict__ srcE, const float* __restrict__ subE,
                     const float* __restrict__ ca, const float* __restrict__ cb,
                     float* __restrict__ dists, int D, int W) {
  __shared__ float As[2][TM * LDT];
  __shared__ float Bs[2][TN * LDT];
  __shared__ float diag[TM + TN - 1];

  const int tid     = threadIdx.x;
  const int lane    = tid & 31;
  const int wv      = tid >> 5;
  const int rowBase = blockIdx.x * TM;
  const int colBase = blockIdx.y * TN;

  for (int i = tid; i < TM + TN - 1; i += 256) diag[i] = 0.0f;

  // 8 waves: 4 along M (32 rows each = 2 tiles), 2 along N (64 cols each = 4 tiles)
  const int m0w = (wv & 3) * 32;
  const int n0w = (wv >> 2) * 64;
  const int ml  = lane & 15;
  const int kb  = (lane >> 4) << 1;   // K pair selected by lane half

  const v8f zero8 = {0.f,0.f,0.f,0.f,0.f,0.f,0.f,0.f};
  v8f acc[2][4];
#pragma unroll
  for (int i = 0; i < 2; ++i)
#pragma unroll
    for (int j = 0; j < 4; ++j) acc[i][j] = zero8;

  // register staging for one K-chunk (8 float4 per thread)
  float4 ra[4], rb[4];
  auto load_regs = [&](int kc) {
#pragma unroll
    for (int i = 0; i < 4; ++i) {
      int s   = tid + i * 256;
      int row = s >> 3;
      int c4  = (s & 7) << 2;
      ra[i] = *(const float4*)(srcE + (size_t)(rowBase + row) * D + kc + c4);
      rb[i] = *(const float4*)(subE + (size_t)(colBase + row) * D + kc + c4);
    }
  };
  auto store_lds = [&](int buf) {
#pragma unroll
    for (int i = 0; i < 4; ++i) {
      int s   = tid + i * 256;
      int row = s >> 3;
      int c4  = (s & 7) << 2;
      *(float4*)&As[buf][row * LDT + c4] = ra[i];
      *(float4*)&Bs[buf][row * LDT + c4] = rb[i];
    }
  };

  const int nc = D / KC;
  load_regs(0);
  store_lds(0);
  int buf = 0;
  for (int c = 0; c < nc; ++c) {
    __syncthreads();                       // current LDS buffer ready; the other buffer free
    if (c + 1 < nc) load_regs((c + 1) * KC);   // issue next chunk's global loads
#pragma unroll
    for (int kk = 0; kk < KC; kk += 4) {
      v2f a[2], b[4];
#pragma unroll
      for (int mt = 0; mt < 2; ++mt)
        a[mt] = *(const v2f*)&As[buf][(m0w + mt * 16 + ml) * LDT + kk + kb];
#pragma unroll
      for (int nt = 0; nt < 4; ++nt)
        b[nt] = *(const v2f*)&Bs[buf][(n0w + nt * 16 + ml) * LDT + kk + kb];
#pragma unroll
      for (int mt = 0; mt < 2; ++mt)
#pragma unroll
        for (int nt = 0; nt < 4; ++nt)
          acc[mt][nt] = __builtin_amdgcn_wmma_f32_16x16x4_f32(
              false, a[mt], false, b[nt], (short)0, acc[mt][nt], false, false);
    }
    if (c + 1 < nc) store_lds(buf ^ 1);    // drain loads, fill other buffer
    buf ^= 1;
  }

  __syncthreads();
  // epilogue: within each lane half, the 16 lanes of a given acc element map to distinct diagonals
  const int dBase = rowBase - colBase - (TN - 1);
  const int mhalf = (lane >> 4) << 3;   // +8 rows for lanes 16..31
#pragma unroll
  for (int mt = 0; mt < 2; ++mt) {
#pragma unroll
    for (int nt = 0; nt < 4; ++nt) {
      const int n     = n0w + nt * 16 + ml;
      const int j     = colBase + n;
      const float cbv = cb[j];
#pragma unroll
      for (int v = 0; v < 8; ++v) {
        const int r = rowBase + m0w + mt * 16 + v + mhalf;
        const int d = r - j;
        float sq = ca[r] + cbv - 2.0f * acc[mt][nt][v];
        float rn = sqrtf(fmaxf(sq, 0.0f));
        if (d >= 0 && d < W) atomicAdd(&diag[d - dBase], rn);  // ds_add_f32
      }
    }
  }
  __syncthreads();
  for (int i = tid; i < TM + TN - 1; i += 256) {
    int d = dBase + i;
    if (d >= 0 && d < W) {
      float vsum = diag[i];
      if (vsum != 0.0f) atomicAdd(&dists[d], vsum);
    }
  }
}

// ------------------------------------------------------------------- argmin
__global__ __launch_bounds__(1024)
void argmin_kernel(const float* __restrict__ dists, int W, int* __restrict__ sel) {
  __shared__ float sv[1024];
  __shared__ int   si[1024];
  const int tid = threadIdx.x;
  float best = INFINITY; int bi = 0;
  for (int i = tid; i < W; i += 1024) {
    float v = dists[i];
    if (v < best) { best = v; bi = i; }
  }
  sv[tid] = best; si[tid] = bi;
  __syncthreads();
  for (int s = 512; s > 0; s >>= 1) {
    if (tid < s) {
      float v2 = sv[tid + s]; int i2 = si[tid + s];
      if (v2 < sv[tid] || (v2 == sv[tid] && i2 < si[tid])) { sv[tid] = v2; si[tid] = i2; }
    }
    __syncthreads();
  }
  if (tid == 0) *sel = si[0];
}

// ------------------------------------------- Kabsch (means, cov, SVD, RMSD)
__global__ __launch_bounds__(256)
void finalize_kernel(const float* __restrict__ srcC, const float* __restrict__ subC,
                     const int* __restrict__ selp, int L, float* __restrict__ out) {
  __shared__ float red[256];
  __shared__ float bc[20];
  const int tid = threadIdx.x;
  const int sel = *selp;
  const float* w = srcC + (size_t)sel * 3;

  // means of window and subject coords
  float a6[6] = {0,0,0,0,0,0};
  for (int i = tid; i < L; i += 256) {
    a6[0] += w[i*3+0];    a6[1] += w[i*3+1];    a6[2] += w[i*3+2];
    a6[3] += subC[i*3+0]; a6[4] += subC[i*3+1]; a6[5] += subC[i*3+2];
  }
  for (int c = 0; c < 6; ++c) {
    __syncthreads();
    red[tid] = a6[c];
    __syncthreads();
    for (int s = 128; s > 0; s >>= 1) { if (tid < s) red[tid] += red[tid + s]; __syncthreads(); }
    if (tid == 0) bc[c] = red[0] / (float)L;
  }
  __syncthreads();
  const float am0 = bc[0], am1 = bc[1], am2 = bc[2];
  const float bm0 = bc[3], bm1 = bc[4], bm2 = bc[5];

  // 3x3 covariance
  float cv[9] = {0,0,0,0,0,0,0,0,0};
  for (int i = tid; i < L; i += 256) {
    float a0 = w[i*3+0]-am0, a1 = w[i*3+1]-am1, a2 = w[i*3+2]-am2;
    float b0 = subC[i*3+0]-bm0, b1 = subC[i*3+1]-bm1, b2 = subC[i*3+2]-bm2;
    cv[0]+=a0*b0; cv[1]+=a0*b1; cv[2]+=a0*b2;
    cv[3]+=a1*b0; cv[4]+=a1*b1; cv[5]+=a1*b2;
    cv[6]+=a2*b0; cv[7]+=a2*b1; cv[8]+=a2*b2;
  }
  for (int c = 0; c < 9; ++c) {
    __syncthreads();
    red[tid] = cv[c];
    __syncthreads();
    for (int s = 128; s > 0; s >>= 1) { if (tid < s) red[tid] += red[tid + s]; __syncthreads(); }
    if (tid == 0) bc[6 + c] = red[0];
  }
  __syncthreads();

  if (tid == 0) {
    // one-sided Jacobi SVD of 3x3: B = cov * V_accum, columns -> orthogonal
    float B[3][3], V[3][3], U[3][3];
    for (int i = 0; i < 3; ++i)
      for (int j = 0; j < 3; ++j) { B[i][j] = bc[6 + i*3 + j]; V[i][j] = (i == j) ? 1.f : 0.f; }
    for (int sweep = 0; sweep < 30; ++sweep) {
      for (int p = 0; p < 2; ++p) {
        for (int q = p + 1; q < 3; ++q) {
          float al = 0.f, be = 0.f, ga = 0.f;
          for (int i = 0; i < 3; ++i) {
            al += B[i][p]*B[i][p]; be += B[i][q]*B[i][q]; ga += B[i][p]*B[i][q];
          }
          if (ga == 0.f || fabsf(ga) <= 1e-12f * sqrtf(al * be)) continue;
          float zeta = (be - al) / (2.0f * ga);
          float t    = copysignf(1.0f, zeta) / (fabsf(zeta) + sqrtf(1.0f + zeta*zeta));
          float cc   = 1.0f / sqrtf(1.0f + t*t);
          float sn   = cc * t;
          for (int i = 0; i < 3; ++i) {
            float bp = B[i][p], bq = B[i][q];
            B[i][p] = cc*bp - sn*bq; B[i][q] = sn*bp + cc*bq;
            float vp = V[i][p], vq = V[i][q];
            V[i][p] = cc*vp - sn*vq; V[i][q] = sn*vp + cc*vq;
          }
        }
      }
    }
    for (int k = 0; k < 3; ++k) {
      float nrm = sqrtf(B[0][k]*B[0][k] + B[1][k]*B[1][k] + B[2][k]*B[2][k]);
      if (nrm > 1e-30f) { U[0][k]=B[0][k]/nrm; U[1][k]=B[1][k]/nrm; U[2][k]=B[2][k]/nrm; }
      else { U[0][k] = (k==0); U[1][k] = (k==1); U[2][k] = (k==2); }
    }
    // R = V * U^T; reference's "flip row 2 of v" == flip row 2 of R if det<0
    float R[3][3];
    for (int i = 0; i < 3; ++i)
      for (int j = 0; j < 3; ++j)
        R[i][j] = V[i][0]*U[j][0] + V[i][1]*U[j][1] + V[i][2]*U[j][2];
    float det = R[0][0]*(R[1][1]*R[2][2]-R[1][2]*R[2][1])
              - R[0][1]*(R[1][0]*R[2][2]-R[1][2]*R[2][0])
              + R[0][2]*(R[1][0]*R[2][1]-R[1][1]*R[2][0]);
    if (det < 0.f) { R[2][0] = -R[2][0]; R[2][1] = -R[2][1]; R[2][2] = -R[2][2]; }
    bc[6]=R[0][0];  bc[7]=R[0][1];  bc[8]=R[0][2];
    bc[9]=R[1][0];  bc[10]=R[1][1]; bc[11]=R[1][2];
    bc[12]=R[2][0]; bc[13]=R[2][1]; bc[14]=R[2][2];
    bc[15] = bm0 - (R[0][0]*am0 + R[0][1]*am1 + R[0][2]*am2);
    bc[16] = bm1 - (R[1][0]*am0 + R[1][1]*am1 + R[1][2]*am2);
    bc[17] = bm2 - (R[2][0]*am0 + R[2][1]*am1 + R[2][2]*am2);
  }
  __syncthreads();

  const float R00=bc[6],R01=bc[7],R02=bc[8],R10=bc[9],R11=bc[10],R12=bc[11],
              R20=bc[12],R21=bc[13],R22=bc[14],t0=bc[15],t1=bc[16],t2=bc[17];
  float se = 0.f;
  for (int i = tid; i < L; i += 256) {
    float x = w[i*3+0], y = w[i*3+1], z = w[i*3+2];
    float e0 = R00*x + R01*y + R02*z + t0 - subC[i*3+0];
    float e1 = R10*x + R11*y + R12*z + t1 - subC[i*3+1];
    float e2 = R20*x + R21*y + R22*z + t2 - subC[i*3+2];
    se += e0*e0 + e1*e1 + e2*e2;
  }
  __syncthreads();
  red[tid] = se;
  __syncthreads();
  for (int s = 128; s > 0; s >>= 1) { if (tid < s) red[tid] += red[tid + s]; __syncthreads(); }
  if (tid == 0) {
    out[0] = sqrtf(red[0] / (float)L);
    for (int i = 0; i < 9; ++i) out[1 + i] = bc[6 + i];
    out[10] = bc[15]; out[11] = bc[16]; out[12] = bc[17];
    out[13] = (float)sel;
  }
}

// ------------------------------------------------------------------ launch
extern "C" void kernel_launch(void* const* d_in, const int* in_sizes, int n_in,
                              void* d_out, int out_size, void* d_ws, size_t ws_size,
                              hipStream_t stream) {
  (void)n_in; (void)out_size; (void)ws_size;
  const float* srcE = (const float*)d_in[0];
  const float* subE = (const float*)d_in[1];
  const float* srcC = (const float*)d_in[2];
  const float* subC = (const float*)d_in[3];
  const int N = in_sizes[2] / 3;        // 8192
  const int L = in_sizes[3] / 3;        // 2048
  const int D = in_sizes[0] / N;        // 512
  const int W = N - L + 1;              // 6145

  float* ws    = (float*)d_ws;
  float* dists = ws;
  size_t off   = (size_t)((W + 63) & ~63);
  float* ca    = ws + off; off += (size_t)N;
  float* cb    = ws + off; off += (size_t)L;
  int*   sel   = (int*)(ws + off);

  zero_kernel<<<(W + 255) / 256, 256, 0, stream>>>(dists, W);
  row_stats_kernel<<<(N + 7) / 8, 256, 0, stream>>>(srcE, ca, N, D,  1.0f, 0.0f);
  row_stats_kernel<<<(L + 7) / 8, 256, 0, stream>>>(subE, cb, L, D, -1.0f, (float)D * EPSV * EPSV);
  dim3 grid(N / TM, L / TN);
  dot_diag_kernel<<<grid, 256, 0, stream>>>(srcE, subE, ca, cb, dists, D, W);
  argmin_kernel<<<1, 1024, 0, stream>>>(dists, W, sel);
  finalize_kernel<<<1, 256, 0, stream>>>(srcC, subC, sel, L, (float*)d_out);
}